// ComboTopoLoss_26302379721190
// MI455X (gfx1250) — compile-verified
//
#include <hip/hip_runtime.h>
#include <cstdint>
#include <cstddef>

// ---------------------------------------------------------------------------
// ComboTopoLoss for MI455X (gfx1250).
//   labels, predictions: (16,1,1024,1024) fp32.  Output: scalar fp32.
// Memory-bound streaming reduction: CDNA5 async global->LDS loads
// (ASYNCcnt pipeline, depth 3) + ds_load_b128 + wave32 shuffle reductions.
// Softplus term uses native v_exp_f32/v_log_f32 so VALU stays under the
// 23.3 TB/s HBM roofline (~5.6 us for 128 MiB).
// Deterministic: no floating-point atomics; fixed-order tree reductions.
// ---------------------------------------------------------------------------

#define WAIT_ASYNC(n) asm volatile("s_wait_asynccnt " #n ::: "memory")

__device__ __forceinline__ void async_load_b128_to_lds(uint32_t lds_addr,
                                                       const void* gaddr) {
  // GV mode: 64-bit per-lane address in a VGPR pair, LDS byte address in VGPR.
  asm volatile("global_load_async_to_lds_b128 %0, %1, off"
               :
               : "v"(lds_addr), "v"((unsigned long long)(uintptr_t)gaddr)
               : "memory");
}

#define NPATCH 1024     // 16 images * 8*8 patches
#define PSTAGES 16      // 16 stages * 8 rows/stage = 128 rows
#define NBUF 4          // LDS buffers (pipeline depth 3 < NBUF, avoids LDS WAR)

__global__ __launch_bounds__(256) void topo_patch_kernel(
    const float* __restrict__ lab, const float* __restrict__ prd,
    float* __restrict__ ws) {
  __shared__ float4 sL[NBUF][256];
  __shared__ float4 sP[NBUF][256];
  __shared__ float red[4][8];

  const int tid  = threadIdx.x;
  const int lane = tid & 31;
  const int wave = tid >> 5;

  const int patch = blockIdx.x;          // 0..1023
  const int b  = patch >> 6;             // image
  const int ph = (patch >> 3) & 7;       // patch row
  const int pw = patch & 7;              // patch col
  const size_t base = (size_t)b * 1048576u + (size_t)ph * 131072u + (size_t)pw * 128u;
  const float* lab0 = lab + base + (size_t)lane * 4u;   // lane's float4 in a row
  const float* prd0 = prd + base + (size_t)lane * 4u;

  // stage s: wave w streams row (s*8 + w); 32 lanes cover 128 floats (b128 each)
  auto issue = [&](int s) {
    const int buf = s & (NBUF - 1);
    const size_t off = (size_t)(((s << 3) + wave) * 1024u);
    async_load_b128_to_lds((uint32_t)(uintptr_t)&sL[buf][tid], lab0 + off);
    async_load_b128_to_lds((uint32_t)(uintptr_t)&sP[buf][tid], prd0 + off);
  };

  float bce  = 0.0f;
  float lsum = 0.0f;                       // labels are {0,1}: exact in fp32
  float xmin = 3.0e38f, xmax = -3.0e38f;   // raw predictions

  auto process = [&](int s) {
    const int buf = s & (NBUF - 1);
    const float4 lv = sL[buf][tid];   // ds_load_b128; wave reads only its own slots
    const float4 pv = sP[buf][tid];
    const float l[4] = {lv.x, lv.y, lv.z, lv.w};
    const float x[4] = {pv.x, pv.y, pv.z, pv.w};
#pragma unroll
    for (int j = 0; j < 4; ++j) {
      // BCE-with-logits: max(x,0) - x*l + log1p(exp(-|x|)).
      // exp(-|x|) in (0,1] -> log(1+e) is cancellation-free; native trans ops.
      const float sp = __logf(1.0f + __expf(-fabsf(x[j])));
      bce += fmaxf(x[j], 0.0f) - x[j] * l[j] + sp;
      lsum += l[j];
      xmin = fminf(xmin, x[j]);
      xmax = fmaxf(xmax, x[j]);
    }
  };

  // --- software pipeline: depth 3 (3 stages = 6 async ops in flight/wave) ---
  issue(0); issue(1); issue(2);
  for (int s = 0; s < PSTAGES - 3; ++s) {
    WAIT_ASYNC(4);          // <=2 stages outstanding => stage s complete
    process(s);
    issue(s + 3);
  }
  WAIT_ASYNC(4); process(13);
  WAIT_ASYNC(2); process(14);
  WAIT_ASYNC(0); process(15);

  // --- wave32 reduction (fixed order -> deterministic) ---
#pragma unroll
  for (int m = 16; m >= 1; m >>= 1) {
    bce  += __shfl_xor(bce, m, 32);
    lsum += __shfl_xor(lsum, m, 32);
    xmin = fminf(xmin, __shfl_xor(xmin, m, 32));
    xmax = fmaxf(xmax, __shfl_xor(xmax, m, 32));
  }
  if (lane == 0) {
    red[0][wave] = bce;  red[1][wave] = lsum;
    red[2][wave] = xmin; red[3][wave] = xmax;
  }
  __syncthreads();

  if (tid == 0) {
    float B = 0.0f, Ls = 0.0f;
    float Xmin = 3.0e38f, Xmax = -3.0e38f;
#pragma unroll
    for (int w = 0; w < 8; ++w) {
      B  += red[0][w];
      Ls += red[1][w];
      Xmin = fminf(Xmin, red[2][w]);
      Xmax = fmaxf(Xmax, red[3][w]);
    }
    ws[patch] = B;  // per-patch BCE partial sum

    // gt = 1-label: valid <=> patch contains both 0s and 1s.
    const bool valid = (Ls > 0.5f) && (Ls < 16383.5f);
    // lh = sigmoid(1-pred) is decreasing in pred:
    const float pmax_lh = 1.0f / (1.0f + expf(-(1.0f - Xmin)));
    const float pmin_lh = 1.0f / (1.0f + expf(-(1.0f - Xmax)));
    const float pl = (pmax_lh - 1.0f) * (pmax_lh - 1.0f) + pmin_lh * pmin_lh;
    ws[NPATCH + patch] = valid ? pl : 0.0f;
  }
}

__global__ __launch_bounds__(256) void topo_final_kernel(
    const float* __restrict__ ws, float* __restrict__ out) {
  __shared__ double sb[256];
  __shared__ double st[256];
  const int t = threadIdx.x;
  double b  = (double)ws[t] + (double)ws[t + 256] + (double)ws[t + 512] +
              (double)ws[t + 768];
  double tp = (double)ws[NPATCH + t] + (double)ws[NPATCH + t + 256] +
              (double)ws[NPATCH + t + 512] + (double)ws[NPATCH + t + 768];
  sb[t] = b; st[t] = tp;
  __syncthreads();
#pragma unroll
  for (int m = 128; m >= 1; m >>= 1) {
    if (t < m) { sb[t] += sb[t + m]; st[t] += st[t + m]; }
    __syncthreads();
  }
  if (t == 0) {
    // bce mean over 16*1024*1024 elems, topo mean over 16 samples
    out[0] = (float)(sb[0] * (1.0 / 16777216.0) + st[0] * (1.0 / 16.0));
  }
}

extern "C" void kernel_launch(void* const* d_in, const int* in_sizes, int n_in,
                              void* d_out, int out_size, void* d_ws, size_t ws_size,
                              hipStream_t stream) {
  (void)in_sizes; (void)n_in; (void)out_size; (void)ws_size;
  const float* labels = (const float*)d_in[0];
  const float* preds  = (const float*)d_in[1];
  float* out = (float*)d_out;
  float* ws  = (float*)d_ws;   // [0,1024): bce partials, [1024,2048): topo partials

  topo_patch_kernel<<<dim3(NPATCH), dim3(256), 0, stream>>>(labels, preds, ws);
  topo_final_kernel<<<dim3(1), dim3(256), 0, stream>>>(ws, out);
}